// MultiQueryAttention_17927193493793
// MI455X (gfx1250) — compile-verified
//
#include <hip/hip_runtime.h>
#include <math.h>

typedef __attribute__((ext_vector_type(2))) float v2f;
typedef __attribute__((ext_vector_type(8))) float v8f;
typedef __attribute__((ext_vector_type(4))) unsigned int u32x4;
typedef __attribute__((ext_vector_type(8))) int i32x8;
typedef __attribute__((ext_vector_type(4))) int i32x4;

#define B_   2
#define S_   2048
#define H_   2048
#define NH_  16
#define D_   128
#define MROWS (B_ * S_)            // 4096
#define QCOLS (NH_ * D_)           // 2048
#define SCALE_ 0.08838834764831845f // 128^-0.5

static __device__ __forceinline__ v8f wmma_f32(v2f a, v2f b, v8f c) {
  // D = A(16x4 f32) * B(4x16 f32) + C(16x16 f32), full fp32 precision
  return __builtin_amdgcn_wmma_f32_16x16x4_f32(false, a, false, b, (short)0, c,
                                               false, false);
}

// ---------------------------------------------------------------------------
// TDM: DMA a 2-D fp32 tile (tileX cols x tileY rows, row stride strideElems)
// from global memory into LDS at byte offset ldsOff, inserting a row pad so
// LDS row stride = tileX + 4 floats.  Descriptor per CDNA5 ISA D# layout.
//   padIntervalCode: dwords-per-row = 2<<code  (32 -> 4, 64 -> 5)
//   padAmountCode:   pad dwords = code+1       (4  -> 3)
// Issued by one wave only (TDM ignores EXEC); tracked by TENSORcnt.
// ---------------------------------------------------------------------------
static __device__ __forceinline__ void tdm_load_2d(const void* gaddr,
                                                   unsigned ldsOff, int tileX,
                                                   int tileY, long strideElems,
                                                   int padIntervalCode,
                                                   int padAmountCode) {
  unsigned long long ga = (unsigned long long)(uintptr_t)gaddr;
  u32x4 g0;
  g0.x = 1u;                                  // count=1, load, no gather
  g0.y = ldsOff;                              // lds_addr (bytes)
  g0.z = (unsigned)ga;                        // global_addr[31:0]
  g0.w = (unsigned)(ga >> 32) | (2u << 30);   // global_addr[56:32], type=2
  const unsigned td0 = 1u << 30, td1 = 1u << 30;  // huge tensor dims: no OOB
  i32x8 g1;
  g1[0] = (2 << 16)                     // data_size = 4 bytes
          | (1 << 20)                   // pad_enable
          | (padIntervalCode << 22) | (padAmountCode << 25);
  g1[1] = (int)((td0 & 0xFFFFu) << 16);                      // tensor_dim0 lo
  g1[2] = (int)(((td0 >> 16) & 0xFFFFu) | ((td1 & 0xFFFFu) << 16));
  g1[3] = (int)(((td1 >> 16) & 0xFFFFu) | ((unsigned)tileX << 16));
  g1[4] = tileY & 0xFFFF;                                    // tile_dim1/2
  unsigned long long st0 = (unsigned long long)strideElems;  // dim0 stride
  g1[5] = (int)(st0 & 0xFFFFFFFFu);
  g1[6] = (int)((st0 >> 32) & 0xFFFFu);                      // stride1 = 0
  g1[7] = 0;
  i32x4 z4 = {0, 0, 0, 0};
  i32x8 z8 = {0, 0, 0, 0, 0, 0, 0, 0};
  // 6-arg form (clang-23 / therock-10.0 headers): groups 0..3, extra, cpol
  __builtin_amdgcn_tensor_load_to_lds(g0, g1, z4, z4, z8, 0);
}

// ---------------------------------------------------------------------------
// Kernel 1: fused QKV projection.  C[4096 x 2304] = hs[4096x2048] x [Wq|Wk|Wv]
// Block: 256 threads (8 waves), 64x64 tile, K-step 32, TDM double-buffered.
// ---------------------------------------------------------------------------
__global__ __launch_bounds__(256) void qkv_gemm_kernel(
    const float* __restrict__ hs, const float* __restrict__ Wq,
    const float* __restrict__ Wk, const float* __restrict__ Wv,
    float* __restrict__ Qb, float* __restrict__ Kb, float* __restrict__ Vb) {
  __shared__ float ldsA[2][64 * 36];  // 64 x 32, TDM-padded to stride 36
  __shared__ float ldsB[2][32 * 68];  // 32 x 64, TDM-padded to stride 68

  const int t = threadIdx.x;
  const int lane = t & 31;
  const int w = t >> 5;
  const int l = lane & 15;
  const int hi = lane >> 4;
  const int mSub = (w >> 1) * 16;
  const int nSub = (w & 1) * 32;
  const int mBase = blockIdx.x * 64;
  const int nBase = blockIdx.y * 64;

  const float* Bp;
  float* Cp;
  int nOff, ldb;
  if (nBase < QCOLS) {            Bp = Wq; Cp = Qb; nOff = nBase;              ldb = QCOLS; }
  else if (nBase < QCOLS + D_) {  Bp = Wk; Cp = Kb; nOff = nBase - QCOLS;      ldb = D_; }
  else {                          Bp = Wv; Cp = Vb; nOff = nBase - QCOLS - D_; ldb = D_; }

  v8f acc0 = {0, 0, 0, 0, 0, 0, 0, 0};
  v8f acc1 = {0, 0, 0, 0, 0, 0, 0, 0};

  const int nK = H_ / 32;
  auto issueTile = [&](int i) {
    const int kB = i * 32, bi = i & 1;
    tdm_load_2d(hs + (size_t)mBase * H_ + kB,
                (unsigned)(uintptr_t)(void*)&ldsA[bi][0], 32, 64, H_, 4, 3);
    tdm_load_2d(Bp + (size_t)kB * ldb + nOff,
                (unsigned)(uintptr_t)(void*)&ldsB[bi][0], 64, 32, ldb, 5, 3);
  };
  if (t < 32) issueTile(0);

  for (int i = 0; i < nK; ++i) {
    if (t < 32) {
      if (i + 1 < nK) {
        issueTile(i + 1);
        __builtin_amdgcn_s_wait_tensorcnt(2);  // tile i landed (in-order)
      } else {
        __builtin_amdgcn_s_wait_tensorcnt(0);
      }
    }
    __syncthreads();
    const float* lA = ldsA[i & 1];
    const float* lB = ldsB[i & 1];
#pragma unroll
    for (int kk = 0; kk < 8; ++kk) {
      const int kr = kk * 4 + 2 * hi;
      v2f a = *(const v2f*)&lA[(mSub + l) * 36 + kr];
      v2f b0, b1;
      b0.x = lB[kr * 68 + nSub + l];
      b0.y = lB[(kr + 1) * 68 + nSub + l];
      b1.x = lB[kr * 68 + nSub + 16 + l];
      b1.y = lB[(kr + 1) * 68 + nSub + 16 + l];
      acc0 = wmma_f32(a, b0, acc0);
      acc1 = wmma_f32(a, b1, acc1);
    }
    __syncthreads();
  }
#pragma unroll
  for (int v = 0; v < 8; ++v) {
    const size_t row = (size_t)(mBase + mSub + v + 8 * hi);
    Cp[row * ldb + nOff + nSub + l] = acc0[v];
    Cp[row * ldb + nOff + nSub + 16 + l] = acc1[v];
  }
}

// ---------------------------------------------------------------------------
// Kernel 2: RoPE on Q and K (in place).  One thread per (d, d+64) pair.
// ---------------------------------------------------------------------------
__global__ void rope_kernel(float* __restrict__ Qb, float* __restrict__ Kb,
                            const float* __restrict__ cosb,
                            const float* __restrict__ sinb) {
  const int nQ = MROWS * NH_ * (D_ / 2);
  const int nK = MROWS * (D_ / 2);
  int i = blockIdx.x * blockDim.x + threadIdx.x;
  if (i < nQ) {
    int d1 = i & 63;
    int r = i >> 6;
    int h = r & (NH_ - 1);
    int r2 = r >> 4;
    int s = r2 & (S_ - 1);
    int b = r2 >> 11;
    size_t base = (size_t)(b * S_ + s) * QCOLS + h * D_;
    float c1 = cosb[s * D_ + d1], s1 = sinb[s * D_ + d1];
    float c2 = cosb[s * D_ + d1 + 64], s2 = sinb[s * D_ + d1 + 64];
    float x1 = Qb[base + d1], x2 = Qb[base + d1 + 64];
    Qb[base + d1] = x1 * c1 - x2 * s1;
    Qb[base + d1 + 64] = x2 * c2 + x1 * s2;
  } else if (i < nQ + nK) {
    int j = i - nQ;
    int d1 = j & 63;
    int r = j >> 6;
    int s = r & (S_ - 1);
    int b = r >> 11;
    size_t base = (size_t)(b * S_ + s) * D_;
    float c1 = cosb[s * D_ + d1], s1 = sinb[s * D_ + d1];
    float c2 = cosb[s * D_ + d1 + 64], s2 = sinb[s * D_ + d1 + 64];
    float x1 = Kb[base + d1], x2 = Kb[base + d1 + 64];
    Kb[base + d1] = x1 * c1 - x2 * s1;
    Kb[base + d1 + 64] = x2 * c2 + x1 * s2;
  }
}

// ---------------------------------------------------------------------------
// Kernel 3: causal flash attention (multi-query: one shared K/V head).
// Grid (S/64, NH, B); block = 128 (4 waves), each wave owns 16 q-rows.
// ---------------------------------------------------------------------------
__global__ __launch_bounds__(128) void attn_kernel(
    const float* __restrict__ Qb, const float* __restrict__ Kb,
    const float* __restrict__ Vb, float* __restrict__ Ob) {
  __shared__ float ldsP[4][16 * 18];  // per-wave P tile, stride 18

  const int t = threadIdx.x;
  const int lane = t & 31;
  const int w = t >> 5;
  const int l = lane & 15;
  const int hi = lane >> 4;
  const int qTile = blockIdx.x;
  const int h = blockIdx.y;
  const int b = blockIdx.z;
  const int qbase = qTile * 64 + w * 16;
  const size_t rowBase = (size_t)b * S_;

  v2f qa[32];
#pragma unroll
  for (int kk = 0; kk < 32; ++kk) {
    const float* p =
        Qb + (rowBase + qbase + l) * (size_t)QCOLS + h * D_ + kk * 4 + 2 * hi;
    v2f tmp;
    tmp.x = p[0] * SCALE_;
    tmp.y = p[1] * SCALE_;
    qa[kk] = tmp;
  }

  float m_i[8], l_i[8];
  v8f o[8];
#pragma unroll
  for (int v = 0; v < 8; ++v) {
    m_i[v] = -3.0e38f;
    l_i[v] = 0.0f;
    o[v] = (v8f){0, 0, 0, 0, 0, 0, 0, 0};
  }

  const int numKT = qTile * 4 + 4;  // causal bound, uniform across the block
  for (int kt = 0; kt < numKT; ++kt) {
    const int kb = kt * 16;
    v8f s_acc = {0, 0, 0, 0, 0, 0, 0, 0};
    const float* kp = Kb + (rowBase + kb + l) * (size_t)D_;
#pragma unroll
    for (int kk = 0; kk < 32; ++kk) {
      const int kr = kk * 4 + 2 * hi;
      v2f bfr;
      bfr.x = kp[kr];
      bfr.y = kp[kr + 1];
      s_acc = wmma_f32(qa[kk], bfr, s_acc);
    }
#pragma unroll
    for (int v = 0; v < 8; ++v) {
      const int qrow = qbase + v + 8 * hi;
      const int kcol = kb + l;
      float sc = s_acc[v] + ((kcol > qrow) ? -1000000000.0f : 0.0f);
      float r = sc;
      r = fmaxf(r, __shfl_xor(r, 1));
      r = fmaxf(r, __shfl_xor(r, 2));
      r = fmaxf(r, __shfl_xor(r, 4));
      r = fmaxf(r, __shfl_xor(r, 8));
      const float mnew = fmaxf(m_i[v], r);
      const float alpha = __expf(m_i[v] - mnew);
      const float pv = __expf(sc - mnew);
      float rs = pv;
      rs += __shfl_xor(rs, 1);
      rs += __shfl_xor(rs, 2);
      rs += __shfl_xor(rs, 4);
      rs += __shfl_xor(rs, 8);
      l_i[v] = l_i[v] * alpha + rs;
      m_i[v] = mnew;
#pragma unroll
      for (int j = 0; j < 8; ++j) o[j][v] *= alpha;
      ldsP[w][(v + 8 * hi) * 18 + l] = pv;
    }
    __syncthreads();
#pragma unroll
    for (int kk2 = 0; kk2 < 4; ++kk2) {
      const int k0 = kk2 * 4;
      v2f pa = *(const v2f*)&ldsP[w][l * 18 + k0 + 2 * hi];
      const float* vp0 = Vb + (rowBase + kb + k0 + 2 * hi) * (size_t)D_;
      const float* vp1 = vp0 + D_;
#pragma unroll
      for (int j = 0; j < 8; ++j) {
        v2f bfr;
        bfr.x = vp0[j * 16 + l];
        bfr.y = vp1[j * 16 + l];
        o[j] = wmma_f32(pa, bfr, o[j]);
      }
    }
    __syncthreads();
  }
#pragma unroll
  for (int v = 0; v < 8; ++v) {
    const float inv = 1.0f / l_i[v];
    const size_t row = rowBase + qbase + v + 8 * hi;
    float* op = Ob + row * (size_t)QCOLS + h * D_;
#pragma unroll
    for (int j = 0; j < 8; ++j) op[j * 16 + l] = o[j][v] * inv;
  }
}

// ---------------------------------------------------------------------------
// Kernel 4: output projection.  out[4096x2048] = Ob[4096x2048] x Wo[2048x2048]
// TDM double-buffered like kernel 1.
// ---------------------------------------------------------------------------
__global__ __launch_bounds__(256) void gemm_kernel(
    const float* __restrict__ A, const float* __restrict__ Bm,
    float* __restrict__ C, int Kdim, int N) {
  __shared__ float ldsA[2][64 * 36];
  __shared__ float ldsB[2][32 * 68];
  const int t = threadIdx.x;
  const int lane = t & 31;
  const int w = t >> 5;
  const int l = lane & 15;
  const int hi = lane >> 4;
  const int mSub = (w >> 1) * 16;
  const int nSub = (w & 1) * 32;
  const int mBase = blockIdx.x * 64;
  const int nBase = blockIdx.y * 64;

  v8f acc0 = {0, 0, 0, 0, 0, 0, 0, 0};
  v8f acc1 = {0, 0, 0, 0, 0, 0, 0, 0};

  const int nK = Kdim / 32;
  auto issueTile = [&](int i) {
    const int kB = i * 32, bi = i & 1;
    tdm_load_2d(A + (size_t)mBase * Kdim + kB,
                (unsigned)(uintptr_t)(void*)&ldsA[bi][0], 32, 64, Kdim, 4, 3);
    tdm_load_2d(Bm + (size_t)kB * N + nBase,
                (unsigned)(uintptr_t)(void*)&ldsB[bi][0], 64, 32, N, 5, 3);
  };
  if (t < 32) issueTile(0);

  for (int i = 0; i < nK; ++i) {
    if (t < 32) {
      if (i + 1 < nK) {
        issueTile(i + 1);
        __builtin_amdgcn_s_wait_tensorcnt(2);
      } else {
        __builtin_amdgcn_s_wait_tensorcnt(0);
      }
    }
    __syncthreads();
    const float* lA = ldsA[i & 1];
    const float* lB = ldsB[i & 1];
#pragma unroll
    for (int kk = 0; kk < 8; ++kk) {
      const int kr = kk * 4 + 2 * hi;
      v2f a = *(const v2f*)&lA[(mSub + l) * 36 + kr];
      v2f b0, b1;
      b0.x = lB[kr * 68 + nSub + l];
      b0.y = lB[(kr + 1) * 68 + nSub + l];
      b1.x = lB[kr * 68 + nSub + 16 + l];
      b1.y = lB[(kr + 1) * 68 + nSub + 16 + l];
      acc0 = wmma_f32(a, b0, acc0);
      acc1 = wmma_f32(a, b1, acc1);
    }
    __syncthreads();
  }
#pragma unroll
  for (int v = 0; v < 8; ++v) {
    const size_t row = (size_t)(mBase + mSub + v + 8 * hi);
    C[row * N + nBase + nSub + l] = acc0[v];
    C[row * N + nBase + nSub + 16 + l] = acc1[v];
  }
}

// ---------------------------------------------------------------------------
extern "C" void kernel_launch(void* const* d_in, const int* in_sizes, int n_in,
                              void* d_out, int out_size, void* d_ws,
                              size_t ws_size, hipStream_t stream) {
  const float* hs = (const float*)d_in[0];
  // d_in[1] = attention_mask: skipped, causal mask regenerated in-register
  const float* cosb = (const float*)d_in[2];
  const float* sinb = (const float*)d_in[3];
  const float* Wq = (const float*)d_in[4];
  const float* Wk = (const float*)d_in[5];
  const float* Wv = (const float*)d_in[6];
  const float* Wo = (const float*)d_in[7];
  float* out = (float*)d_out;

  float* ws = (float*)d_ws;
  float* Qb = ws;                            // 4096*2048 f32
  float* Kb = Qb + (size_t)MROWS * QCOLS;    // 4096*128
  float* Vb = Kb + (size_t)MROWS * D_;       // 4096*128
  float* Ob = Vb + (size_t)MROWS * D_;       // 4096*2048

  qkv_gemm_kernel<<<dim3(MROWS / 64, (QCOLS + 2 * D_) / 64), 256, 0, stream>>>(
      hs, Wq, Wk, Wv, Qb, Kb, Vb);

  const int nPairs = MROWS * NH_ * (D_ / 2) + MROWS * (D_ / 2);
  rope_kernel<<<(nPairs + 255) / 256, 256, 0, stream>>>(Qb, Kb, cosb, sinb);

  attn_kernel<<<dim3(S_ / 64, NH_, B_), 128, 0, stream>>>(Qb, Kb, Vb, Ob);

  gemm_kernel<<<dim3(MROWS / 64, H_ / 64), 256, 0, stream>>>(Ob, Wo, out, H_,
                                                             H_);
}